// WholeModel_970662609571
// MI455X (gfx1250) — compile-verified
//
#include <hip/hip_runtime.h>
#include <math.h>

typedef float v2f __attribute__((ext_vector_type(2)));
typedef float v8f __attribute__((ext_vector_type(8)));

#define TT 256
#define BB 2048
#define HH 10
#define DD 2

__device__ __forceinline__ v8f zero8() {
  v8f z;
#pragma unroll
  for (int i = 0; i < 8; ++i) z[i] = 0.0f;
  return z;
}

// D = A(16x4 f32) * B(4x16 f32) + C(16x16 f32), exact fp32 WMMA
__device__ __forceinline__ v8f wmma4(v2f a, v2f b, v8f c) {
  return __builtin_amdgcn_wmma_f32_16x16x4_f32(false, a, false, b, (short)0, c,
                                               false, false);
}

__device__ __forceinline__ v8f mm3(const v2f* A, const v2f* B) {
  v8f c = zero8();
  c = wmma4(A[0], B[0], c);
  c = wmma4(A[1], B[1], c);
  c = wmma4(A[2], B[2], c);
  return c;
}

// ---- branch-free fast math ----
__device__ __forceinline__ float frcp(float x) { return __builtin_amdgcn_rcpf(x); }
__device__ __forceinline__ float fexp2(float x) { return __builtin_amdgcn_exp2f(x); }
#define TWO_LOG2E 2.8853900817779268f
#define LOG2E     1.4426950408889634f

#if __has_builtin(__builtin_amdgcn_tanhf)
// gfx1250 hardware v_tanh_f32: single TRANS op per activation
__device__ __forceinline__ float ftanh(float x) {
  return __builtin_amdgcn_tanhf(x);
}
// sigmoid(x)+eps = fma(0.5, tanh(x/2), 0.5+eps)
__device__ __forceinline__ float fsig_eps(float x, float eps) {
  return __builtin_fmaf(0.5f, __builtin_amdgcn_tanhf(0.5f * x), 0.5f + eps);
}
#else
__device__ __forceinline__ float ftanh(float x) {
  return 1.0f - 2.0f * frcp(fexp2(x * TWO_LOG2E) + 1.0f);
}
__device__ __forceinline__ float fsig_eps(float x, float eps) {
  return frcp(1.0f + fexp2(x * -LOG2E)) + eps;
}
#endif

// lane-half swap as pure VALU: v_permlanex16 with identity lane selects.
// Lane i (i<16) reads lane 16+i; lane i (i>=16) reads lane i-16.
__device__ __forceinline__ float sx16(float v) {
  int i = __builtin_bit_cast(int, v);
  int r = __builtin_amdgcn_permlanex16(i, i, 0x76543210, 0xfedcba98, false, false);
  return __builtin_bit_cast(float, r);
}

// Element of augmented W^T: A[m][k] = W[k+row_off][m] for k<K_in,
// bias[m] at k==bias_row, else 0. Rows m>=N_out are zero padding.
__device__ __forceinline__ float wval(const float* W, const float* bias,
                                      int K_in, int N_out, int bias_row,
                                      int k, int m, int row_off) {
  if (m >= N_out) return 0.0f;
  if (k < K_in)   return W[(size_t)(k + row_off) * N_out + m];
  if (k == bias_row && bias) return bias[m];
  return 0.0f;
}

// One K-chunk of W^T in the 16x4 f32 A-matrix layout:
// lanes 0-15 hold {K=4c, 4c+1} of row M=lane; lanes 16-31 hold {4c+2, 4c+3}.
__device__ __forceinline__ v2f achunk(const float* W, const float* bias,
                                      int K_in, int N_out, int bias_row,
                                      int c, int lane, int row_off) {
  int m = lane & 15;
  int k = 4 * c + ((lane >> 4) << 1);
  v2f r;
  r.x = wval(W, bias, K_in, N_out, bias_row, k,     m, row_off);
  r.y = wval(W, bias, K_in, N_out, bias_row, k + 1, m, row_off);
  return r;
}

// Fused generator head: mu^T in rows M=0..1, sg^T in rows M=2..3 (+bias row 10)
__device__ __forceinline__ float gmsval(const float* Wm, const float* bm,
                                        const float* Ws, const float* bs,
                                        int k, int m) {
  if (m >= 4) return 0.0f;
  const float* W = (m < 2) ? Wm : Ws;
  const float* b = (m < 2) ? bm : bs;
  int col = m & 1;
  if (k < 10)  return W[k * DD + col];
  if (k == 10) return b[col];
  return 0.0f;
}
__device__ __forceinline__ v2f gmschunk(const float* Wm, const float* bm,
                                        const float* Ws, const float* bs,
                                        int c, int lane) {
  int m = lane & 15;
  int k = 4 * c + ((lane >> 4) << 1);
  v2f r;
  r.x = gmsval(Wm, bm, Ws, bs, k,     m);
  r.y = gmsval(Wm, bm, Ws, bs, k + 1, m);
  return r;
}

// Convert activation in C layout (lane=batch col, VGPR v = hidden rows v / v+8)
// into three B-matrix K-chunks (rows 0..9 data, row 10 = 1.0 bias, row 11 = 0).
__device__ __forceinline__ void c2b(const v8f h, bool lo, v2f* b) {
  float s0 = sx16(h[0]), s1 = sx16(h[1]);
  float s2 = sx16(h[2]), s3 = sx16(h[3]);
  float s6 = sx16(h[6]), s7 = sx16(h[7]);
  b[0].x = lo ? h[0] : s2;   b[0].y = lo ? h[1] : s3;   // K = 0..3
  b[1].x = lo ? h[4] : s6;   b[1].y = lo ? h[5] : s7;   // K = 4..7
  b[2].x = lo ? s0 : 1.0f;   b[2].y = lo ? s1 : 0.0f;   // K = 8,9 | 10(bias),11
}

__device__ __forceinline__ void tanh8(v8f& c) {
#pragma unroll
  for (int i = 0; i < 8; ++i) c[i] = ftanh(c[i]);
}
__device__ __forceinline__ void sig8(v8f& c, float eps) {
#pragma unroll
  for (int i = 0; i < 8; ++i) c[i] = fsig_eps(c[i], eps);
}
__device__ __forceinline__ void relu8(v8f& c) {
#pragma unroll
  for (int i = 0; i < 8; ++i) c[i] = fmaxf(c[i], 0.0f);
}
// fused mu/sg head: only rows 0..3 are consumed
__device__ __forceinline__ void sig4(v8f& c) {
#pragma unroll
  for (int i = 0; i < 4; ++i) c[i] = fsig_eps(c[i], 0.0f);
}

// x_t as B-matrix chunk (rows: x0, x1, 1, 0). Branchless: all lanes load
// (upper lanes hit their partner's address), constants selected for upper.
__device__ __forceinline__ v2f load_x(const float* seq, int t, int gb, bool lo) {
  v2f r = *(const v2f*)(seq + ((size_t)t * BB + gb) * DD);
  v2f o;
  o.x = lo ? r.x : 1.0f;
  o.y = lo ? r.y : 0.0f;
  return o;
}

// eps in C layout: lower lanes need h0..7, upper lanes need h8,h9.
// Branchless: upper lanes load at +2-float shift so {h8,h9} is the last pair
// loaded (stays in bounds), then two selects. Unused upper rows carry garbage
// that is provably discarded downstream.
__device__ __forceinline__ v8f load_eps(const float* noise, int t, int gb, bool lo) {
  const float* p = noise + ((size_t)t * BB + gb) * HH + (lo ? 0 : 2);
  const v2f* p2 = (const v2f*)p;
  v2f a = p2[0], b = p2[1], c = p2[2], d = p2[3];
  v8f e;
  e[0] = lo ? a.x : d.x;   // lower: h0, upper: h8
  e[1] = lo ? a.y : d.y;   // lower: h1, upper: h9
  e[2] = b.x; e[3] = b.y;
  e[4] = c.x; e[5] = c.y;
  e[6] = d.x; e[7] = d.y;
  return e;
}

extern "C" __global__ void __launch_bounds__(32)
dvbf_scan(const float* __restrict__ seq, const float* __restrict__ noise,
  const float* fi_in_W, const float* fi_in_b, const float* fi_h_W, const float* fi_h_b,
  const float* fi_mu_W, const float* fi_mu_b, const float* fi_sg_W, const float* fi_sg_b,
  const float* inf_in_W, const float* inf_in_b, const float* inf_j_W, const float* inf_j_b,
  const float* inf_h_W, const float* inf_h_b, const float* inf_mu_W, const float* inf_mu_b,
  const float* inf_sg_W, const float* inf_sg_b,
  const float* tr_h_W, const float* tr_h_b, const float* tr_mu_W, const float* tr_mu_b,
  const float* tr_sg_W, const float* tr_sg_b,
  const float* gen_h1_W, const float* gen_h1_b, const float* gen_h2_W, const float* gen_h2_b,
  const float* gen_mu_W, const float* gen_mu_b, const float* gen_sg_W, const float* gen_sg_b,
  float* __restrict__ gens, float* __restrict__ loss_out)
{
  const int  lane = (int)(threadIdx.x & 31u);
  const bool lo   = lane < 16;
  const int  gb   = blockIdx.x * 16 + (lane & 15);
  const float LOG_PI = 1.1447298858494002f;

  // ---- persistent weights in A layout (bias folded into K-row 10 / 2) ----
  v2f A_inf_in = achunk(inf_in_W, inf_in_b, 2, HH, 2, 0, lane, 0);
  v2f A_ijt[3], A_ijb[3], A_ih[3], A_imu[3], A_isg[3];
  v2f A_th[3], A_tmu[3], A_tsg[3];
  v2f A_g1[3], A_g2[3], A_gms[3];
#pragma unroll
  for (int c = 0; c < 3; ++c) {
    A_ijt[c] = achunk(inf_j_W, inf_j_b, 10, HH, 10, c, lane, 0);   // e-part + bias
    A_ijb[c] = achunk(inf_j_W, nullptr, 10, HH, -1, c, lane, 10);  // z-part, no bias
    A_ih[c]  = achunk(inf_h_W,  inf_h_b,  10, HH, 10, c, lane, 0);
    A_imu[c] = achunk(inf_mu_W, inf_mu_b, 10, HH, 10, c, lane, 0);
    A_isg[c] = achunk(inf_sg_W, inf_sg_b, 10, HH, 10, c, lane, 0);
    A_th[c]  = achunk(tr_h_W,   tr_h_b,   10, HH, 10, c, lane, 0);
    A_tmu[c] = achunk(tr_mu_W,  tr_mu_b,  10, HH, 10, c, lane, 0);
    A_tsg[c] = achunk(tr_sg_W,  tr_sg_b,  10, HH, 10, c, lane, 0);
    A_g1[c]  = achunk(gen_h1_W, gen_h1_b, 10, HH, 10, c, lane, 0);
    A_g2[c]  = achunk(gen_h2_W, gen_h2_b, 10, HH, 10, c, lane, 0);
    A_gms[c] = gmschunk(gen_mu_W, gen_mu_b, gen_sg_W, gen_sg_b, c, lane);
  }

  float acc = 0.0f;   // per-lane partial of summed (kl + ll), constants folded out
  v2f bz[3];          // current z in B layout (K rows 0..9 = z, row 10 = 1)

  // ---------------- t = 0 : first_inference ----------------
  {
    v2f A_fi = achunk(fi_in_W, fi_in_b, 2, HH, 2, 0, lane, 0);
    v2f A_fh[3], A_fmu[3], A_fsg[3];
#pragma unroll
    for (int c = 0; c < 3; ++c) {
      A_fh[c]  = achunk(fi_h_W,  fi_h_b,  10, HH, 10, c, lane, 0);
      A_fmu[c] = achunk(fi_mu_W, fi_mu_b, 10, HH, 10, c, lane, 0);
      A_fsg[c] = achunk(fi_sg_W, fi_sg_b, 10, HH, 10, c, lane, 0);
    }
    v2f bx = load_x(seq, 0, gb, lo);
    v8f ep = load_eps(noise, 0, gb, lo);

    v8f h = zero8(); h = wmma4(A_fi, bx, h); tanh8(h);
    v2f bh[3]; c2b(h, lo, bh);
    v8f h2 = mm3(A_fh, bh); tanh8(h2); c2b(h2, lo, bh);
    v8f qmu = mm3(A_fmu, bh); tanh8(qmu);
    v8f qsg = mm3(A_fsg, bh); sig8(qsg, 0.001f);

    // KL(q || N(0,I)) partial: 0.5*(-log prod(s) + sum s + sum mu^2)
    float prodq = 1.0f, ssum = 0.0f;
    int nrows = lo ? 8 : 2;
#pragma unroll
    for (int v = 0; v < 8; ++v) {
      float tadd = __builtin_fmaf(qmu[v], qmu[v], qsg[v]);
      prodq *= (v < nrows) ? qsg[v] : 1.0f;
      ssum  += (v < nrows) ? tadd    : 0.0f;
    }
    acc += 0.5f * (ssum - __logf(prodq));

    v8f z;
#pragma unroll
    for (int v = 0; v < 8; ++v) z[v] = qmu[v] + qsg[v] * ep[v];
    c2b(z, lo, bz);

    // generator(z1), fused mu/sg head
    v8f g = mm3(A_g1, bz); relu8(g);
    v2f bg[3]; c2b(g, lo, bg);
    v8f g2 = mm3(A_g2, bg); relu8(g2); c2b(g2, lo, bg);
    v8f gms = mm3(A_gms, bg); sig4(gms);

    if (lo) {
      float r0 = frcp(gms[2]), r1 = frcp(gms[3]);
      float d0 = bx.x - gms[0], d1 = bx.y - gms[1];
      float a = __logf(gms[2] * gms[3]);
      float b = d0 * d0 * r0 + d1 * d1 * r1;
      acc += -0.5f * (a + b);
      v2f o; o.x = gms[0]; o.y = gms[1];
      *(v2f*)(gens + (size_t)gb * DD) = o;
    }
  }

  // ---------------- t = 1..T-1 : sequential scan ----------------
#pragma unroll 1
  for (int t = 1; t < TT; ++t) {
    // stream in this step's x and eps first (overlap with transition WMMAs)
    v2f bx = load_x(seq, t, gb, lo);
    v8f ep = load_eps(noise, t, gb, lo);

    // transition(z_prev)
    v8f th = mm3(A_th, bz); tanh8(th);
    v2f bt[3]; c2b(th, lo, bt);
    v8f pmu = mm3(A_tmu, bt); tanh8(pmu);
    v8f psg = mm3(A_tsg, bt); tanh8(psg);

    // inference(x_t, z_prev)
    v8f e = zero8(); e = wmma4(A_inf_in, bx, e); tanh8(e);
    v2f be[3]; c2b(e, lo, be);
    v8f jh = zero8();
#pragma unroll
    for (int c = 0; c < 3; ++c) jh = wmma4(A_ijt[c], be[c], jh);
#pragma unroll
    for (int c = 0; c < 3; ++c) jh = wmma4(A_ijb[c], bz[c], jh);
    tanh8(jh);
    v2f bj[3]; c2b(jh, lo, bj);
    v8f ih = mm3(A_ih, bj); tanh8(ih);
    v2f bi[3]; c2b(ih, lo, bi);
    v8f qmu = mm3(A_imu, bi); tanh8(qmu);
    v8f qsg = mm3(A_isg, bi); sig8(qsg, 0.001f);

    // KLD(q || p) partial: sum of logs collapsed into one log of a product
    float prodr = 1.0f, ssum = 0.0f;
    int nrows = lo ? 8 : 2;
#pragma unroll
    for (int v = 0; v < 8; ++v) {
      float rp    = frcp(psg[v]);
      float ratio = psg[v] * frcp(qsg[v]);
      float dm    = pmu[v] - qmu[v];
      float tadd  = __builtin_fmaf(dm, dm, qsg[v]) * rp;
      prodr *= (v < nrows) ? ratio : 1.0f;
      ssum  += (v < nrows) ? tadd  : 0.0f;
    }
    acc += 0.5f * (__logf(prodr) + ssum);

    // z_t = q_mu + q_sg * eps
    v8f z;
#pragma unroll
    for (int v = 0; v < 8; ++v) z[v] = qmu[v] + qsg[v] * ep[v];
    c2b(z, lo, bz);

    // generator(z_t), fused mu/sg head
    v8f g = mm3(A_g1, bz); relu8(g);
    v2f bg[3]; c2b(g, lo, bg);
    v8f g2 = mm3(A_g2, bg); relu8(g2); c2b(g2, lo, bg);
    v8f gms = mm3(A_gms, bg); sig4(gms);

    if (lo) {
      float r0 = frcp(gms[2]), r1 = frcp(gms[3]);
      float d0 = bx.x - gms[0], d1 = bx.y - gms[1];
      float a = __logf(gms[2] * gms[3]);
      float b = d0 * d0 * r0 + d1 * d1 * r1;
      acc += -0.5f * (a + b);
      v2f o; o.x = gms[0]; o.y = gms[1];
      *(v2f*)(gens + ((size_t)t * BB + gb) * DD) = o;
    }
  }

  // fold the per-step constants back in: each of TT steps contributed
  // -0.5*HH (KL dim term) and -LOG_PI (LL const, lower lanes only).
  if (lo) acc += (float)TT * (-0.5f * (float)HH - LOG_PI);

  // combine lane-half partials; output -loss/T
  float tot = acc + sx16(acc);
  if (lo) loss_out[gb] = -tot / (float)TT;
}

extern "C" void kernel_launch(void* const* d_in, const int* in_sizes, int n_in,
                              void* d_out, int out_size, void* d_ws, size_t ws_size,
                              hipStream_t stream) {
  (void)in_sizes; (void)n_in; (void)d_ws; (void)ws_size; (void)out_size;
  const float* p[34];
  for (int i = 0; i < 34; ++i) p[i] = (const float*)d_in[i];
  float* gens = (float*)d_out;
  float* loss = gens + (size_t)TT * BB * DD;
  dvbf_scan<<<BB / 16, 32, 0, stream>>>(
      p[0],  p[1],
      p[2],  p[3],  p[4],  p[5],  p[6],  p[7],  p[8],  p[9],
      p[10], p[11], p[12], p[13], p[14], p[15], p[16], p[17], p[18], p[19],
      p[20], p[21], p[22], p[23], p[24], p[25], p[26], p[27], p[28], p[29],
      p[30], p[31], p[32], p[33],
      gens, loss);
}